// EfficientGlobalPointer_40810779247289
// MI455X (gfx1250) — compile-verified
//
#include <hip/hip_runtime.h>
#include <hip/hip_bf16.h>

typedef __attribute__((ext_vector_type(2))) float v2f;
typedef __attribute__((ext_vector_type(4))) float v4f;
typedef __attribute__((ext_vector_type(8))) float v8f;

#define BATCH   2
#define L_SEQ   2048
#define H_DIM   1024
#define PROJ    128      // 2*HEAD_SIZE
#define HEADS   12
#define ROWS    (BATCH * L_SEQ)   // 4096
#define SPAD    132      // padded LDS stride for score tile (bank-conflict free)

// ---------------------------------------------------------------------------
// Kernel 1: seq_out = inputs @ Wp + bp   (4096x1024 @ 1024x128)
// Block = 16 rows x 128 cols, 8 waves; each wave owns a 16x16 tile.
// A (shared by all 8 waves) is staged in LDS in 16x64 k-chunks; WMMA A
// operands then come from ds_load_b64 instead of 8x-redundant VMEM.
// ---------------------------------------------------------------------------
__global__ __launch_bounds__(256) void proj_gemm_kernel(
    const float* __restrict__ inp, const float* __restrict__ Wp,
    const float* __restrict__ bp, float* __restrict__ seq_out) {
  __shared__ float sA[16 * 64];

  const int tid  = threadIdx.x;
  const int wave = tid >> 5;
  const int lane = tid & 31;
  const int half = lane >> 4;   // lane-half selects K pair
  const int lm   = lane & 15;   // row (A) / col (B,C)
  const int row0 = blockIdx.x * 16;
  const int col0 = wave * 16;

  v8f acc = {};
  for (int kc = 0; kc < H_DIM; kc += 64) {
    __syncthreads();
    {  // cooperative 16x64 A-chunk load: 256 threads x 1 float4
      const int i   = tid * 4;            // 0..1023
      const int r   = i >> 6;             // row in chunk
      const int c   = i & 63;             // col in chunk
      *(v4f*)(sA + i) =
          *(const v4f*)(inp + (size_t)(row0 + r) * H_DIM + kc + c);
    }
    __syncthreads();
#pragma unroll 4
    for (int k0 = 0; k0 < 64; k0 += 4) {
      const int ka = k0 + 2 * half;
      // A: lane lm holds row lm, K = ka, ka+1 (from LDS)
      v2f a = *(const v2f*)(sA + lm * 64 + ka);
      // B: lane lm holds col lm, K = ka, ka+1 (Wp is small & L2-resident)
      v2f b;
      b[0] = Wp[(size_t)(kc + ka) * PROJ + col0 + lm];
      b[1] = Wp[(size_t)(kc + ka + 1) * PROJ + col0 + lm];
      acc = __builtin_amdgcn_wmma_f32_16x16x4_f32(false, a, false, b,
                                                  (short)0, acc, false, false);
    }
  }
  const float bias = bp[col0 + lm];
#pragma unroll
  for (int j = 0; j < 8; ++j) {
    const int m = row0 + j + 8 * half;           // C layout: row j+8*half
    seq_out[(size_t)m * PROJ + col0 + lm] = acc[j] + bias;
  }
}

// ---------------------------------------------------------------------------
// Kernel 2: RoPE(q), RoPE(k) and bias = (seq_out @ Wq + bq)/2.
// One 128-thread block per sequence row; row staged in LDS.
// bias[..., 2h+0] -> biasA (added along m), bias[..., 2h+1] -> biasB (along n)
// ---------------------------------------------------------------------------
__global__ __launch_bounds__(128) void rope_bias_kernel(
    const float* __restrict__ seq_out, const float* __restrict__ Wq,
    const float* __restrict__ bq, float* __restrict__ qrot,
    float* __restrict__ krot, float* __restrict__ biasA,
    float* __restrict__ biasB) {
  __shared__ float s[PROJ];
  const int row = blockIdx.x;
  const int t   = threadIdx.x;                 // 0..127
  s[t] = seq_out[(size_t)row * PROJ + t];
  __syncthreads();

  const int l  = row & (L_SEQ - 1);            // position within sequence
  const int he = t & 63;                       // element within head dim
  const int i2 = he & ~1;                      // 2*(he/2)
  const float ang = (float)l * __powf(10000.0f, -(float)i2 * (1.0f / 64.0f));
  float sn, cs;
  __sincosf(ang, &sn, &cs);
  const float* src = s + ((t < 64) ? 0 : 64);
  float v;
  if ((he & 1) == 0) v = src[he] * cs - src[he + 1] * sn;
  else               v = src[he] * cs + src[he - 1] * sn;
  if (t < 64) qrot[(size_t)row * 64 + he] = v;
  else        krot[(size_t)row * 64 + he] = v;

  if (t < 2 * HEADS) {                         // 24 dots of length 128
    float d = bq[t];
    for (int dd = 0; dd < PROJ; ++dd) d += s[dd] * Wq[dd * (2 * HEADS) + t];
    d *= 0.5f;
    const int h = t >> 1;
    if ((t & 1) == 0) biasA[(size_t)row * HEADS + h] = d;
    else              biasB[(size_t)row * HEADS + h] = d;
  }
}

// ---------------------------------------------------------------------------
// Kernel 3: fused scores + bias broadcast + mask + causal penalty.
// Block = 16 m-rows x 128 n-cols for one batch. Wave w computes the 16x16
// score tile at n-offset 16w via 16 fp32 WMMAs (K=64, q tile from LDS), dumps
// the scaled tile into padded LDS, then the whole block streams
// 12 heads x 16x128 as coalesced non-temporal float4 stores (512B/wave-instr,
// NT so the 403MB stream doesn't evict the ~5MB hot set from L2).
// ---------------------------------------------------------------------------
__global__ __launch_bounds__(256) void logits_kernel(
    const float* __restrict__ qrot, const float* __restrict__ krot,
    const float* __restrict__ biasA, const float* __restrict__ biasB,
    const int* __restrict__ mask, float* __restrict__ out) {
  __shared__ float sQ[16 * 64];
  __shared__ float sS[16 * SPAD];
  __shared__ float sBA[16 * HEADS];
  __shared__ float sBB[128 * HEADS];
  __shared__ int   sMm[16];
  __shared__ int   sMn[128];

  const int b   = blockIdx.z;
  const int m0  = blockIdx.y * 16;
  const int nb  = blockIdx.x * 128;
  const int tid = threadIdx.x;

  {  // cooperative 16x64 q-tile load: 256 threads x 1 float4
    const int i = tid * 4;
    const int r = i >> 6;
    const int c = i & 63;
    *(v4f*)(sQ + i) =
        *(const v4f*)(qrot + (size_t)(b * L_SEQ + m0 + r) * 64 + c);
  }
  for (int i = tid; i < 16 * HEADS; i += 256)
    sBA[i] = biasA[((size_t)(b * L_SEQ + m0 + i / HEADS)) * HEADS + (i % HEADS)];
  for (int i = tid; i < 128 * HEADS; i += 256)
    sBB[i] = biasB[((size_t)(b * L_SEQ + nb + i / HEADS)) * HEADS + (i % HEADS)];
  if (tid < 16)  sMm[tid] = mask[b * L_SEQ + m0 + tid];
  if (tid < 128) sMn[tid] = mask[b * L_SEQ + nb + tid];
  __syncthreads();

  const int wave = tid >> 5;
  const int lane = tid & 31;
  const int half = lane >> 4;
  const int lm   = lane & 15;
  const int nloc = wave * 16 + lm;

  const float* kbase = krot + (size_t)(b * L_SEQ + nb + nloc) * 64;
  v8f acc = {};
#pragma unroll
  for (int k0 = 0; k0 < 64; k0 += 4) {
    const int ka = k0 + 2 * half;
    v2f a  = *(const v2f*)(sQ + lm * 64 + ka);   // A = q tile (16x64) from LDS
    v2f kb = *(const v2f*)(kbase + ka);          // B = k^T: col n holds k-row n
    acc = __builtin_amdgcn_wmma_f32_16x16x4_f32(false, a, false, kb,
                                                (short)0, acc, false, false);
  }
  // dump scaled score tile to LDS (padded stride -> lane halves hit
  // different banks)
#pragma unroll
  for (int j = 0; j < 8; ++j)
    sS[(j + 8 * half) * SPAD + nloc] = acc[j] * 0.125f;   // 1/sqrt(64)
  __syncthreads();

  // streaming epilogue: 512 float4 groups per head, 256 threads -> 2 iters
  for (int idx = tid; idx < 16 * 32; idx += 256) {
    const int r  = idx >> 5;           // local m row
    const int nl = (idx & 31) * 4;     // local n (float4 aligned)
    const int m  = m0 + r;
    const v4f s  = *(const v4f*)(sS + r * SPAD + nl);
    float base[4];
    bool  okc[4];
    const int mm = sMm[r];
#pragma unroll
    for (int c = 0; c < 4; ++c) {
      const int n = nb + nl + c;
      base[c] = s[c] - ((m > n) ? 1.0e12f : 0.0f);   // strict-lower-tri penalty
      okc[c]  = (mm != 0) && (sMn[nl + c] != 0);
    }
    float* op = out + (((size_t)b * HEADS) * L_SEQ + m) * L_SEQ + nb + nl;
#pragma unroll
    for (int h = 0; h < HEADS; ++h) {
      const float ba = sBA[r * HEADS + h];
      v4f o;
#pragma unroll
      for (int c = 0; c < 4; ++c) {
        const float v = base[c] + ba + sBB[(nl + c) * HEADS + h];
        o[c] = okc[c] ? v : -__builtin_inff();
      }
      __builtin_nontemporal_store(o, (v4f*)(op + (size_t)h * L_SEQ * L_SEQ));
    }
  }
}

// ---------------------------------------------------------------------------
extern "C" void kernel_launch(void* const* d_in, const int* in_sizes, int n_in,
                              void* d_out, int out_size, void* d_ws,
                              size_t ws_size, hipStream_t stream) {
  const float* inp  = (const float*)d_in[0];   // (2,2048,1024) f32
  const int*   mask = (const int*)d_in[1];     // (2,2048) i32
  const float* Wp   = (const float*)d_in[2];   // (1024,128) f32
  const float* bp   = (const float*)d_in[3];   // (128,) f32
  const float* Wq   = (const float*)d_in[4];   // (128,24) f32
  const float* bq   = (const float*)d_in[5];   // (24,) f32
  float* out = (float*)d_out;                  // (2,12,2048,2048) f32

  float* ws      = (float*)d_ws;
  float* seq_out = ws;                               // 4096*128
  float* qrot    = seq_out + (size_t)ROWS * PROJ;    // 4096*64
  float* krot    = qrot + (size_t)ROWS * 64;         // 4096*64
  float* biasA   = krot + (size_t)ROWS * 64;         // 4096*12
  float* biasB   = biasA + (size_t)ROWS * HEADS;     // 4096*12

  proj_gemm_kernel<<<ROWS / 16, 256, 0, stream>>>(inp, Wp, bp, seq_out);
  rope_bias_kernel<<<ROWS, 128, 0, stream>>>(seq_out, Wq, bq, qrot, krot,
                                             biasA, biasB);
  logits_kernel<<<dim3(L_SEQ / 128, L_SEQ / 16, BATCH), 256, 0, stream>>>(
      qrot, krot, biasA, biasB, mask, out);
}